// AttentionHawkes_1915555414162
// MI455X (gfx1250) — compile-verified
//
#include <hip/hip_runtime.h>
#include <math.h>

#define B_     32
#define D_     2048
#define L_     2048
#define TWOD_  4096
#define LCHUNK_ 8                 // l-chunks for the mix_sum partial pass
#define LSUB_  (L_ / LCHUNK_)     // 256 rows per chunk

typedef __attribute__((ext_vector_type(2))) float v2f;
typedef __attribute__((ext_vector_type(8))) float v8f;

// ---------------------------------------------------------------------------
// K1: q[b,e] = sum_d query[b,d] * W_in[e,d]   (M=32, N=2048, K=2048, f32 WMMA)
// One wave per 16x16 output tile; V_WMMA_F32_16X16X4_F32 K-loop.
// A (16x4 f32): lanes 0-15 -> M=lane, K=0,1 ; lanes 16-31 -> K=2,3.
// B (4x16 f32): lanes 0-15 -> N=lane, K=0,1 ; lanes 16-31 -> K=2,3.
// C/D (16x16 f32): VGPR r -> M = r (+8 for lanes 16-31), N = lane&15.
// ---------------------------------------------------------------------------
__global__ __launch_bounds__(32) void k_qgemm(const float* __restrict__ query,
                                              const float* __restrict__ W_in,
                                              float* __restrict__ q_out) {
  const int ntile  = blockIdx.x;       // 0..127
  const int mtile  = blockIdx.y;       // 0..1
  const int lane   = threadIdx.x;
  const int half   = lane >> 4;
  const int lanelo = lane & 15;

  const float* arow = query + (size_t)(mtile * 16 + lanelo) * D_;
  const float* brow = W_in  + (size_t)(ntile * 16 + lanelo) * D_;  // B[k][n]=W_in[n,k]

  v8f c = {};
  for (int k0 = 0; k0 < D_; k0 += 4) {
    const int kb = k0 + half * 2;
    v2f a, b;
    a.x = arow[kb];  a.y = arow[kb + 1];
    b.x = brow[kb];  b.y = brow[kb + 1];
    c = __builtin_amdgcn_wmma_f32_16x16x4_f32(false, a, false, b,
                                              (short)0, c, false, false);
  }
  const int col = ntile * 16 + lanelo;
#pragma unroll
  for (int r = 0; r < 8; ++r) {
    const int row = mtile * 16 + r + half * 8;
    q_out[(size_t)row * D_ + col] = c[r];
  }
}

// ---------------------------------------------------------------------------
// K2: scores[b,l] = q[b,:] . context[b,l,:]   (first 512 MB context stream)
// 8 waves per block, one row per wave, q[b] staged in LDS, b128 loads.
// 8192 blocks -> plenty of concurrency for the HBM stream.
// ---------------------------------------------------------------------------
__global__ __launch_bounds__(256) void k_scores(const float* __restrict__ q,
                                                const float* __restrict__ context,
                                                float* __restrict__ scores) {
  __shared__ float qs[D_];             // 8 KB
  const int b = blockIdx.x;
  const int t = threadIdx.x;
  for (int i = t; i < D_; i += 256) qs[i] = q[(size_t)b * D_ + i];
  __syncthreads();

  const int wave = t >> 5;
  const int lane = t & 31;
  const int l    = blockIdx.y * 8 + wave;

  const float4* crow = (const float4*)(context + ((size_t)b * L_ + l) * D_);
  const float4* q4   = (const float4*)qs;

  float acc = 0.f;
  for (int i = lane; i < D_ / 4; i += 32) {
    float4 cv = crow[i];
    float4 qv = q4[i];
    acc += cv.x * qv.x + cv.y * qv.y + cv.z * qv.z + cv.w * qv.w;
  }
#pragma unroll
  for (int off = 16; off > 0; off >>= 1) acc += __shfl_xor(acc, off, 32);
  if (lane == 0) scores[(size_t)b * L_ + l] = acc;
}

// ---------------------------------------------------------------------------
// K3: softmax over L per batch; emit attn (to d_out + ws) and the Hawkes
// weight w2[b,l] = ae[b] * attn[b,l] * exp(-ab[b]*delta_t[b,l]).
// ---------------------------------------------------------------------------
__global__ __launch_bounds__(256) void k_softmax(const float* __restrict__ scores,
                                                 const float* __restrict__ delta_t,
                                                 const float* __restrict__ ae,
                                                 const float* __restrict__ ab,
                                                 float* __restrict__ attn_ws,
                                                 float* __restrict__ w2_ws,
                                                 float* __restrict__ attn_out) {
  __shared__ float red[256];
  const int b = blockIdx.x;
  const int t = threadIdx.x;

  float v[8];
  float m = -1e30f;
#pragma unroll
  for (int j = 0; j < 8; ++j) {
    v[j] = scores[(size_t)b * L_ + t + j * 256];
    m = fmaxf(m, v[j]);
  }
  red[t] = m; __syncthreads();
  for (int s = 128; s > 0; s >>= 1) {
    if (t < s) red[t] = fmaxf(red[t], red[t + s]);
    __syncthreads();
  }
  m = red[0]; __syncthreads();

  float sum = 0.f;
#pragma unroll
  for (int j = 0; j < 8; ++j) { v[j] = expf(v[j] - m); sum += v[j]; }
  red[t] = sum; __syncthreads();
  for (int s = 128; s > 0; s >>= 1) {
    if (t < s) red[t] += red[t + s];
    __syncthreads();
  }
  const float inv = 1.0f / red[0];
  const float aev = ae[b];
  const float abv = ab[b];
#pragma unroll
  for (int j = 0; j < 8; ++j) {
    const int l = t + j * 256;
    const float a = v[j] * inv;
    attn_ws[(size_t)b * L_ + l]  = a;
    attn_out[(size_t)b * L_ + l] = a;
    const float bt = expf(-abv * delta_t[(size_t)b * L_ + l]);
    w2_ws[(size_t)b * L_ + l] = aev * a * bt;
  }
}

// ---------------------------------------------------------------------------
// K4a: partial mix_sum over an l-chunk (second 512 MB context stream).
//   part[lc][b][d] = sum_{l in chunk lc} attn*c + relu(w2*c)
// Grid (B, 2, LCHUNK_) = 512 blocks; float4 (b128) streaming loads; per-chunk
// attn/w2 staged in LDS; prefetch a few rows ahead (global_prefetch_b8).
// ---------------------------------------------------------------------------
__global__ __launch_bounds__(256) void k_mixsum_part(const float* __restrict__ context,
                                                     const float* __restrict__ attn,
                                                     const float* __restrict__ w2,
                                                     float* __restrict__ part) {
  __shared__ float s_w1[LSUB_];        // 1 KB
  __shared__ float s_w2[LSUB_];        // 1 KB
  const int b     = blockIdx.x;
  const int halfd = blockIdx.y;        // 0..1 -> 1024-wide d slice
  const int lc    = blockIdx.z;        // 0..LCHUNK_-1
  const int t     = threadIdx.x;
  const int l0    = lc * LSUB_;

  s_w1[t] = attn[(size_t)b * L_ + l0 + t];
  s_w2[t] = w2[(size_t)b * L_ + l0 + t];
  __syncthreads();

  const int d = halfd * 1024 + t * 4;
  const float* cb = context + ((size_t)b * L_ + l0) * D_ + d;

  float4 acc = make_float4(0.f, 0.f, 0.f, 0.f);
  for (int l = 0; l < LSUB_; ++l) {
    if (l + 8 < LSUB_) __builtin_prefetch(cb + (size_t)(l + 8) * D_, 0, 1);
    const float4 cv = *(const float4*)(cb + (size_t)l * D_);
    const float a1 = s_w1[l];
    const float a2 = s_w2[l];
    acc.x += a1 * cv.x + fmaxf(a2 * cv.x, 0.f);
    acc.y += a1 * cv.y + fmaxf(a2 * cv.y, 0.f);
    acc.z += a1 * cv.z + fmaxf(a2 * cv.z, 0.f);
    acc.w += a1 * cv.w + fmaxf(a2 * cv.w, 0.f);
  }
  *(float4*)(part + ((size_t)lc * B_ + b) * D_ + d) = acc;
}

// K4b: reduce the LCHUNK_ partials -> mix_sum[b,d]  (tiny: 64K elements)
__global__ __launch_bounds__(256) void k_mixsum_reduce(const float* __restrict__ part,
                                                       float* __restrict__ mix_sum) {
  const int idx = blockIdx.x * 256 + threadIdx.x;   // over B_*D_
  float s = 0.f;
#pragma unroll
  for (int j = 0; j < LCHUNK_; ++j) s += part[(size_t)j * B_ * D_ + idx];
  mix_sum[idx] = s;
}

// ---------------------------------------------------------------------------
// K5: out[b,e] = tanh( sum_c [mix_sum | q][b,c] * W_out[e,c] )
// M=32, N=2048, K=4096 f32 WMMA, same tiling as K1.
// ---------------------------------------------------------------------------
__global__ __launch_bounds__(32) void k_outgemm(const float* __restrict__ mix_sum,
                                                const float* __restrict__ q,
                                                const float* __restrict__ W_out,
                                                float* __restrict__ out) {
  const int ntile  = blockIdx.x;
  const int mtile  = blockIdx.y;
  const int lane   = threadIdx.x;
  const int half   = lane >> 4;
  const int lanelo = lane & 15;

  const float* ms = mix_sum + (size_t)(mtile * 16 + lanelo) * D_;
  const float* qr = q       + (size_t)(mtile * 16 + lanelo) * D_;
  const float* wr = W_out   + (size_t)(ntile * 16 + lanelo) * TWOD_;

  v8f c = {};
  for (int k0 = 0; k0 < TWOD_; k0 += 4) {
    const int kb = k0 + half * 2;
    const float* ap = (kb < D_) ? (ms + kb) : (qr + (kb - D_));
    v2f a, b;
    a.x = ap[0];  a.y = ap[1];
    b.x = wr[kb]; b.y = wr[kb + 1];
    c = __builtin_amdgcn_wmma_f32_16x16x4_f32(false, a, false, b,
                                              (short)0, c, false, false);
  }
  const int col = ntile * 16 + lanelo;
#pragma unroll
  for (int r = 0; r < 8; ++r) {
    const int row = mtile * 16 + r + half * 8;
    out[(size_t)row * D_ + col] = tanhf(c[r]);
  }
}

// ---------------------------------------------------------------------------
extern "C" void kernel_launch(void* const* d_in, const int* in_sizes, int n_in,
                              void* d_out, int out_size, void* d_ws, size_t ws_size,
                              hipStream_t stream) {
  const float* query   = (const float*)d_in[0];   // [B,1,D]
  const float* context = (const float*)d_in[1];   // [B,L,D]
  const float* delta_t = (const float*)d_in[2];   // [B,L]
  const float* W_in    = (const float*)d_in[3];   // [D,D]
  const float* W_out   = (const float*)d_in[4];   // [D,2D]
  const float* ae      = (const float*)d_in[5];   // [B]
  const float* ab      = (const float*)d_in[6];   // [B]

  float* out  = (float*)d_out;                    // [B,D]   (65536)
  float* attn = (float*)d_out + B_ * D_;          // [B,L]   (65536)

  float* ws      = (float*)d_ws;
  float* q_ws    = ws;                                  // B*D
  float* scores  = ws + 1 * B_ * D_;                    // B*L
  float* attn_ws = ws + 2 * B_ * D_;                    // B*L
  float* w2_ws   = ws + 3 * B_ * D_;                    // B*L
  float* part_ws = ws + 4 * B_ * D_;                    // LCHUNK_*B*D (2 MB)
  float* mix_ws  = ws + (4 + LCHUNK_) * B_ * D_;        // B*D  (total ~3.25 MB)

  dim3 gGemm(D_ / 16, B_ / 16);                         // 128 x 2 tiles
  k_qgemm<<<gGemm, 32, 0, stream>>>(query, W_in, q_ws);

  dim3 gScores(B_, L_ / 8);                             // 8192 blocks
  k_scores<<<gScores, 256, 0, stream>>>(q_ws, context, scores);

  k_softmax<<<B_, 256, 0, stream>>>(scores, delta_t, ae, ab,
                                    attn_ws, w2_ws, attn);

  dim3 gMix(B_, 2, LCHUNK_);                            // 512 blocks
  k_mixsum_part<<<gMix, 256, 0, stream>>>(context, attn_ws, w2_ws, part_ws);
  k_mixsum_reduce<<<B_ * D_ / 256, 256, 0, stream>>>(part_ws, mix_ws);

  k_outgemm<<<gGemm, 32, 0, stream>>>(mix_ws, q_ws, W_out, out);
}